// TransR_34737695490087
// MI455X (gfx1250) — compile-verified
//
#include <hip/hip_runtime.h>
#include <hip/hip_bf16.h>

typedef __attribute__((ext_vector_type(2))) float v2f;
typedef __attribute__((ext_vector_type(4))) float v4f;
typedef __attribute__((ext_vector_type(8))) float v8f;

#define E_CNT 500000
#define R_CNT 1000
#define D_DIM 128
#define K_DIM 64
#define B_CNT 16384

#define M_STRIDE 130   // LDS row stride for m (avoid 64-bank conflicts)
#define C_STRIDE 132   // LDS col stride for diff

// ---------------- phase 0: zero relation histogram ----------------
__global__ void zero_counts_kernel(int* counts) {
    int i = blockIdx.x * blockDim.x + threadIdx.x;
    if (i < R_CNT) counts[i] = 0;
}

// ---------------- phase 1: histogram of relation ids ----------------
__global__ void hist_kernel(const int* __restrict__ r, int* __restrict__ counts) {
    int b = blockIdx.x * blockDim.x + threadIdx.x;
    if (b < B_CNT) atomicAdd(&counts[r[b]], 1);
}

// ---------------- phase 2: exclusive scan (single block) ----------------
__global__ __launch_bounds__(1024) void scan_kernel(const int* __restrict__ counts,
                                                    int* __restrict__ offsets,
                                                    int* __restrict__ cursors) {
    __shared__ int buf[1024];
    int tid = threadIdx.x;
    int v = (tid < R_CNT) ? counts[tid] : 0;
    buf[tid] = v;
    __syncthreads();
    // Hillis-Steele inclusive scan
    for (int off = 1; off < 1024; off <<= 1) {
        int x = (tid >= off) ? buf[tid - off] : 0;
        __syncthreads();
        buf[tid] += x;
        __syncthreads();
    }
    if (tid < R_CNT) {
        int excl = buf[tid] - v;
        offsets[tid] = excl;
        cursors[tid] = excl;
    }
}

// ---------------- phase 3: scatter triple ids into relation buckets ----------------
__global__ void scatter_kernel(const int* __restrict__ r, int* __restrict__ cursors,
                               int* __restrict__ sorted) {
    int b = blockIdx.x * blockDim.x + threadIdx.x;
    if (b < B_CNT) {
        int rr = r[b];
        int pos = atomicAdd(&cursors[rr], 1);
        sorted[pos] = b;
    }
}

// ---------------- phase 4: per-relation GEMM via WMMA f32 16x16x4 ----------------
// out(64 x n) = m(64x128) @ (h_e - t_e)(128 x n) + r_e, tiled n in chunks of 16.
__global__ __launch_bounds__(128) void transr_gemm_kernel(
    const int* __restrict__ h, const int* __restrict__ t,
    const float* __restrict__ ent, const float* __restrict__ rel,
    const float* __restrict__ tm,
    const int* __restrict__ counts, const int* __restrict__ offsets,
    const int* __restrict__ sorted, float* __restrict__ out)
{
    __shared__ float m_lds[K_DIM * M_STRIDE];   // 64x(128+pad) f32 ~ 33.3 KB
    __shared__ float diff_lds[16 * C_STRIDE];   // 16 cols x (128+pad)   ~ 8.4 KB
    __shared__ float re_lds[K_DIM];
    __shared__ int   col_b[16];

    const int relid = blockIdx.x;
    const int cnt = counts[relid];
    if (cnt == 0) return;
    const int start = offsets[relid];
    const int tid = threadIdx.x;

    // Stage m[relid] (64x128) into LDS: thread -> (row = tid/2, half-row of 64 floats)
    {
        const float* src = tm + (size_t)relid * (K_DIM * D_DIM);
        const int row = tid >> 1;
        const int half = (tid & 1) * 64;
        const float* srow = src + row * D_DIM + half;
        float* drow = m_lds + row * M_STRIDE + half;
#pragma unroll
        for (int i = 0; i < 64; i += 4)
            *(v4f*)(drow + i) = *(const v4f*)(srow + i);
    }
    if (tid < K_DIM) re_lds[tid] = rel[relid * K_DIM + tid];
    __syncthreads();

    const int wave  = tid >> 5;     // 0..3 -> M rows [16w, 16w+16)
    const int lane  = tid & 31;
    const int lhalf = lane >> 4;    // 0/1 -> K sub-pair / M +8 in C
    const int l16   = lane & 15;

    const int ntiles = (cnt + 15) >> 4;
    for (int tile = 0; tile < ntiles; ++tile) {
        // Gather 16 columns of Diff = h_e - t_e into LDS.
        {
            if (tid < 16)
                col_b[tid] = (tile * 16 + tid < cnt) ? sorted[start + tile * 16 + tid] : -1;
            const int col  = tid >> 3;          // 0..15
            const int dseg = (tid & 7) * 16;    // 16 floats per thread
            const int ci = tile * 16 + col;
            float* dp = diff_lds + col * C_STRIDE + dseg;
            if (ci < cnt) {
                const int b = sorted[start + ci];
                const float* hp = ent + (size_t)h[b] * D_DIM + dseg;
                const float* tp = ent + (size_t)t[b] * D_DIM + dseg;
#pragma unroll
                for (int i = 0; i < 16; i += 4) {
                    v4f hv = *(const v4f*)(hp + i);
                    v4f tv = *(const v4f*)(tp + i);
                    *(v4f*)(dp + i) = hv - tv;
                }
            } else {
                const v4f z = {0.0f, 0.0f, 0.0f, 0.0f};
#pragma unroll
                for (int i = 0; i < 16; i += 4) *(v4f*)(dp + i) = z;
            }
        }
        __syncthreads();

        // 16x16x4 f32 WMMA, K = 128 in 32 steps. EXEC is all-ones here (uniform flow).
        v8f acc = {};
        const float* arow = m_lds    + (wave * 16 + l16) * M_STRIDE + 2 * lhalf;
        const float* brow = diff_lds + l16 * C_STRIDE + 2 * lhalf;
#pragma unroll 8
        for (int k0 = 0; k0 < D_DIM; k0 += 4) {
            v2f a = *(const v2f*)(arow + k0);   // (M=l16+16w, K=k0+2*lhalf .. +1)
            v2f b = *(const v2f*)(brow + k0);   // (K=k0+2*lhalf .. +1, N=l16)
            acc = __builtin_amdgcn_wmma_f32_16x16x4_f32(
                false, a, false, b, (short)0, acc, false, false);
        }

        // C/D layout: element i -> row k = 16*wave + 8*lhalf + i, col = l16.
        {
            const int ci = tile * 16 + l16;
            if (ci < cnt) {
                const int b = col_b[l16];
                const int kbase = wave * 16 + 8 * lhalf;
                float* op = out + (size_t)b * K_DIM + kbase;
#pragma unroll
                for (int i = 0; i < 8; ++i)
                    op[i] = acc[i] + re_lds[kbase + i];
            }
        }
        __syncthreads();   // protect diff_lds / col_b before next tile
    }
}

extern "C" void kernel_launch(void* const* d_in, const int* in_sizes, int n_in,
                              void* d_out, int out_size, void* d_ws, size_t ws_size,
                              hipStream_t stream) {
    const int*   h   = (const int*)d_in[0];
    const int*   t   = (const int*)d_in[1];
    const int*   r   = (const int*)d_in[2];
    const float* ent = (const float*)d_in[3];  // (E, D, 1)
    const float* rel = (const float*)d_in[4];  // (R, K, 1)
    const float* tm  = (const float*)d_in[5];  // (R, K, D)
    float* out = (float*)d_out;                // (B, K)

    int* ws      = (int*)d_ws;
    int* counts  = ws;                 // [R]
    int* offsets = ws + R_CNT;         // [R]
    int* cursors = ws + 2 * R_CNT;     // [R]
    int* sorted  = ws + 3 * R_CNT;     // [B]

    zero_counts_kernel<<<(R_CNT + 255) / 256, 256, 0, stream>>>(counts);
    hist_kernel<<<(B_CNT + 255) / 256, 256, 0, stream>>>(r, counts);
    scan_kernel<<<1, 1024, 0, stream>>>(counts, offsets, cursors);
    scatter_kernel<<<(B_CNT + 255) / 256, 256, 0, stream>>>(r, cursors, sorted);
    transr_gemm_kernel<<<R_CNT, 128, 0, stream>>>(h, t, ent, rel, tm,
                                                  counts, offsets, sorted, out);
}